// EliteLinear_63488206570164
// MI455X (gfx1250) — compile-verified
//
#include <hip/hip_runtime.h>
#include <hip/hip_bf16.h>

// MPO-MPS contraction + truncated-SVD sweep for MI455X (gfx1250).
// fp32 throughout (SVD precision) -> V_WMMA_F32_16X16X4_F32 for the dense GEMMs.

#define LSITES 12
#define DBOND  48
#define DPHYS  2
#define CHI    48
#define NCOL   (CHI * DBOND)     // 2304
#define MROWS  96                // padded SVD row count (max 2*chi_max)
#define KMAX   48
#define NB     (KMAX * DPHYS * DBOND)  // 4608 rows of factored-absorb GEMM

typedef float v2f __attribute__((ext_vector_type(2)));
typedef float v8f __attribute__((ext_vector_type(8)));

// ---------------------------------------------------------------------------
// Site 0 boundary: M(96x2304), rows 0..1 = B[0][e=0,a=0][b][f*48+dd], rest 0.
__global__ void mpo_site0_kernel(const float* __restrict__ mps,
                                 const float* __restrict__ cores,
                                 float* __restrict__ M) {
    int idx = blockIdx.x * blockDim.x + threadIdx.x;
    if (idx >= MROWS * NCOL) return;
    int r = idx / NCOL, j = idx % NCOL;
    float val = 0.0f;
    if (r < DPHYS) {                 // r = b (output physical index)
        int f = j / DBOND, dd = j % DBOND;
        // cores[l=0][a=0][b=r][c][dd] * mps[l=0][e=0][c][f]
        #pragma unroll
        for (int c = 0; c < DPHYS; ++c)
            val += cores[(r * DPHYS + c) * DBOND + dd] * mps[c * CHI + f];
    }
    M[idx] = val;
}

// ---------------------------------------------------------------------------
// Gram: G(96x96) = M(96x2304) @ M^T via V_WMMA_F32_16X16X4_F32.
// One wave per 16x16 C tile; K loop of 576 x (K=4) WMMA steps.
// f32 A-frag layout (ISA 7.12.2): lane%16 = row, K = vgpr + (lane<16 ? 0 : 2).
__global__ void gram_wmma_kernel(const float* __restrict__ M,
                                 float* __restrict__ G) {
    int tile = blockIdx.x;           // 36 tiles (6x6)
    int tr = tile / 6, tc = tile % 6;
    int lane = threadIdx.x;          // 32 threads = 1 wave
    int r0 = tr * 16, c0 = tc * 16;
    int row = lane & 15;
    int khalf = (lane < 16) ? 0 : 2;
    v8f acc = {};
    for (int kk = 0; kk < NCOL; kk += 4) {
        v2f a, b;
        a.x = M[(r0 + row) * NCOL + kk + khalf];
        a.y = M[(r0 + row) * NCOL + kk + khalf + 1];
        // B = M^T : B[K][N] = M[c0+N][kk+K]; B layout mirrors A.
        b.x = M[(c0 + row) * NCOL + kk + khalf];
        b.y = M[(c0 + row) * NCOL + kk + khalf + 1];
        acc = __builtin_amdgcn_wmma_f32_16x16x4_f32(
            false, a, false, b, (short)0, acc, false, false);
    }
    // C layout: VGPR v, lane l: row = v + (l<16 ? 0 : 8), col = l%16
    int half8 = (lane < 16) ? 0 : 8;
    float* ap = (float*)&acc;
    #pragma unroll
    for (int v = 0; v < 8; ++v)
        G[(r0 + v + half8) * MROWS + c0 + (lane & 15)] = ap[v];
}

// ---------------------------------------------------------------------------
// Parallel cyclic Jacobi eigensolver on symmetric G (96x96, LDS), eigenvectors
// V in workspace (global). Tournament pairing: 48 disjoint rotations/round.
// Writes Usort (96x48, top-48 eigencolumns, eigenvalue-descending) and the
// site's output tensor U[:, :k] (m x k row-major).
__global__ void jacobi_eig_kernel(const float* __restrict__ Gin,
                                  float* __restrict__ Vb,
                                  float* __restrict__ Usort,
                                  float* __restrict__ outU,
                                  int m, int k) {
    __shared__ float G[MROWS * MROWS];      // 36 KB
    __shared__ float cs[MROWS / 2], sn[MROWS / 2];
    __shared__ int   pp[MROWS / 2], qq[MROWS / 2];
    __shared__ int   ord[MROWS];
    const int tid = threadIdx.x, nt = blockDim.x;

    for (int i = tid; i < MROWS * MROWS; i += nt) {
        G[i] = Gin[i];
        Vb[i] = ((i / MROWS) == (i % MROWS)) ? 1.0f : 0.0f;
    }
    __syncthreads();

    const int NR = MROWS - 1;                // 95 rounds per sweep
    for (int sweep = 0; sweep < 10; ++sweep) {
        for (int r = 0; r < NR; ++r) {
            if (tid < MROWS / 2) {
                int p, q;
                if (tid == 0) { p = r % NR; q = MROWS - 1; }
                else { p = (r + tid) % NR; q = (r + NR - tid) % NR; }
                if (p > q) { int t = p; p = q; q = t; }
                float app = G[p * MROWS + p];
                float aqq = G[q * MROWS + q];
                float apq = G[p * MROWS + q];
                float c = 1.0f, s = 0.0f;
                if (fabsf(apq) > 1e-18f) {
                    float tau = (aqq - app) / (2.0f * apq);
                    float t = ((tau >= 0.0f) ? 1.0f : -1.0f) /
                              (fabsf(tau) + sqrtf(1.0f + tau * tau));
                    c = 1.0f / sqrtf(1.0f + t * t);
                    s = t * c;
                }
                cs[tid] = c; sn[tid] = s; pp[tid] = p; qq[tid] = q;
            }
            __syncthreads();
            // Row rotations (J^T * G): disjoint row pairs -> parallel.
            for (int w = tid; w < (MROWS / 2) * MROWS; w += nt) {
                int t = w / MROWS, j = w % MROWS;
                int p = pp[t], q = qq[t];
                float c = cs[t], s = sn[t];
                float gp = G[p * MROWS + j], gq = G[q * MROWS + j];
                G[p * MROWS + j] = c * gp - s * gq;
                G[q * MROWS + j] = s * gp + c * gq;
            }
            __syncthreads();
            // Column rotations (G * J) and eigenvector accumulation V <- V*J.
            for (int w = tid; w < (MROWS / 2) * MROWS; w += nt) {
                int t = w / MROWS, i = w % MROWS;
                int p = pp[t], q = qq[t];
                float c = cs[t], s = sn[t];
                float gp = G[i * MROWS + p], gq = G[i * MROWS + q];
                G[i * MROWS + p] = c * gp - s * gq;
                G[i * MROWS + q] = s * gp + c * gq;
                float vp = Vb[i * MROWS + p], vq = Vb[i * MROWS + q];
                Vb[i * MROWS + p] = c * vp - s * vq;
                Vb[i * MROWS + q] = s * vp + c * vq;
            }
            __syncthreads();
        }
    }
    // Top-48 eigenvalue selection (descending), serial on thread 0 (96 items).
    if (tid == 0) {
        for (int i = 0; i < MROWS; ++i) ord[i] = i;
        for (int i = 0; i < KMAX; ++i) {
            int best = i;
            for (int j = i + 1; j < MROWS; ++j)
                if (G[ord[j] * MROWS + ord[j]] > G[ord[best] * MROWS + ord[best]])
                    best = j;
            int t = ord[i]; ord[i] = ord[best]; ord[best] = t;
        }
    }
    __syncthreads();
    for (int w = tid; w < MROWS * KMAX; w += nt) {
        int rr = w / KMAX, j = w % KMAX;
        Usort[rr * KMAX + j] = Vb[rr * MROWS + ord[j]];
    }
    // tensors[i] = U[:, :k].reshape(rl, 2, k)  ==  row-major (m x k) copy
    for (int w = tid; w < m * k; w += nt) {
        int rr = w / k, j = w % k;
        outU[w] = Vb[rr * MROWS + ord[j]];
    }
}

// ---------------------------------------------------------------------------
// SV = S*Vh = U_k^T @ M  (48 x 2304); rows >= k zeroed so padded pipeline
// stays exact downstream.
__global__ void sv_kernel(const float* __restrict__ Usort,
                          const float* __restrict__ M,
                          float* __restrict__ SV, int k) {
    int idx = blockIdx.x * blockDim.x + threadIdx.x;
    if (idx >= KMAX * NCOL) return;
    int j = idx / NCOL, col = idx % NCOL;
    float acc = 0.0f;
    if (j < k) {
        for (int r = 0; r < MROWS; ++r)
            acc += Usort[r * KMAX + j] * M[r * NCOL + col];
    }
    SV[idx] = acc;
}

// ---------------------------------------------------------------------------
// Factored absorb stage A: contract the MPO bond 'a'.
// A1[((k'*2+b)*48 + dd)][e*2+c] = sum_a SV[k'][e*48+a] * core[a][b][c][dd]
__global__ void absorb_stageA_kernel(const float* __restrict__ SV,
                                     const float* __restrict__ core,
                                     float* __restrict__ A1) {
    int idx = blockIdx.x * blockDim.x + threadIdx.x;
    if (idx >= NB * MROWS) return;
    int row = idx / MROWS;           // (k'*2+b)*48 + dd
    int q   = idx % MROWS;           // e*2 + c
    int dd = row % DBOND;
    int kb = row / DBOND;            // k'*2 + b
    int b = kb % DPHYS, kp = kb / DPHYS;
    int e = q / DPHYS, c = q % DPHYS;
    const float* svrow = SV + kp * NCOL + e * DBOND;         // SV[k'][e*48 + a]
    const float* cb    = core + b * (DPHYS * DBOND) + c * DBOND + dd; // + a*192
    float acc = 0.0f;
    for (int a = 0; a < DBOND; ++a)
        acc += svrow[a] * cb[a * (DPHYS * DPHYS * DBOND)];
    A1[idx] = acc;
}

// ---------------------------------------------------------------------------
// Factored absorb stage B (WMMA): T(4608 x 48) = A1(4608 x 96) @ mps(96 x 48),
// where mps site memory (chi,d,chi) is already the (e*2+c, f) matrix.
__global__ void absorb_gemmB_kernel(const float* __restrict__ A1,
                                    const float* __restrict__ Bm,
                                    float* __restrict__ T) {
    int tile = blockIdx.x;           // 288 row-tiles x 3 col-tiles = 864
    int tr = tile / 3, tc = tile % 3;
    int lane = threadIdx.x;          // 1 wave
    int r0 = tr * 16, c0 = tc * 16;
    int row = lane & 15;
    int khalf = (lane < 16) ? 0 : 2;
    v8f acc = {};
    for (int kk = 0; kk < MROWS; kk += 4) {
        v2f a, b;
        a.x = A1[(r0 + row) * MROWS + kk + khalf];
        a.y = A1[(r0 + row) * MROWS + kk + khalf + 1];
        b.x = Bm[(kk + khalf) * KMAX + c0 + row];
        b.y = Bm[(kk + khalf + 1) * KMAX + c0 + row];
        acc = __builtin_amdgcn_wmma_f32_16x16x4_f32(
            false, a, false, b, (short)0, acc, false, false);
    }
    int half8 = (lane < 16) ? 0 : 8;
    float* ap = (float*)&acc;
    #pragma unroll
    for (int v = 0; v < 8; ++v)
        T[(r0 + v + half8) * KMAX + c0 + (lane & 15)] = ap[v];
}

// ---------------------------------------------------------------------------
// Scatter T[(k'*2+b)*48+dd][f] -> Mnext[(k'*2+b)][f*48+dd] (full 96x2304 cover)
__global__ void absorb_scatter_kernel(const float* __restrict__ T,
                                      float* __restrict__ Mn) {
    int idx = blockIdx.x * blockDim.x + threadIdx.x;
    if (idx >= NB * KMAX) return;
    int row = idx / KMAX, f = idx % KMAX;
    int dd = row % DBOND, kb = row / DBOND;
    Mn[kb * NCOL + f * DBOND + dd] = T[idx];
}

// ---------------------------------------------------------------------------
// tensors[11] = right-boundary column (f=0, dd=0) of final absorbed M.
__global__ void final_site_kernel(const float* __restrict__ M,
                                  float* __restrict__ out) {
    int r = blockIdx.x * blockDim.x + threadIdx.x;
    if (r < MROWS) out[r] = M[r * NCOL];
}

// ---------------------------------------------------------------------------
extern "C" void kernel_launch(void* const* d_in, const int* in_sizes, int n_in,
                              void* d_out, int out_size, void* d_ws, size_t ws_size,
                              hipStream_t stream) {
    const float* mps   = (const float*)d_in[0];   // (12, 48, 2, 48)
    const float* cores = (const float*)d_in[1];   // (12, 48, 2, 2, 48)
    float* out = (float*)d_out;
    float* ws  = (float*)d_ws;

    float* MA = ws;                          // 96*2304
    float* MB = MA + MROWS * NCOL;           // 96*2304
    float* G  = MB + MROWS * NCOL;           // 96*96
    float* Vb = G  + MROWS * MROWS;          // 96*96
    float* Us = Vb + MROWS * MROWS;          // 96*48
    float* SV = Us + MROWS * KMAX;           // 48*2304
    float* A1 = SV + KMAX * NCOL;            // 4608*96
    float* T  = A1 + NB * MROWS;             // 4608*48
    // total: ~1.23M floats (~4.9 MB)

    // Output tensor offsets (flat, return order); sizes: rl*2*k per site.
    static const int koff[12] = {0, 4, 20, 84, 340, 1364, 4436,
                                 9044, 13652, 18260, 22868, 27476};
    static const int rls[11]  = {1, 2, 4, 8, 16, 32, 48, 48, 48, 48, 48};

    mpo_site0_kernel<<<(MROWS * NCOL + 255) / 256, 256, 0, stream>>>(mps, cores, MA);

    float* Mc = MA;
    float* Mn = MB;
    for (int i = 0; i < 11; ++i) {
        int rl = rls[i];
        int m = 2 * rl;
        int k = (m < KMAX) ? m : KMAX;
        const float* core_n = cores + (size_t)(i + 1) * DBOND * DPHYS * DPHYS * DBOND;
        const float* mps_n  = mps   + (size_t)(i + 1) * CHI * DPHYS * CHI;

        gram_wmma_kernel<<<36, 32, 0, stream>>>(Mc, G);
        jacobi_eig_kernel<<<1, 256, 0, stream>>>(G, Vb, Us, out + koff[i], m, k);
        sv_kernel<<<(KMAX * NCOL + 255) / 256, 256, 0, stream>>>(Us, Mc, SV, k);
        absorb_stageA_kernel<<<(NB * MROWS + 255) / 256, 256, 0, stream>>>(SV, core_n, A1);
        absorb_gemmB_kernel<<<864, 32, 0, stream>>>(A1, mps_n, T);
        absorb_scatter_kernel<<<(NB * KMAX + 255) / 256, 256, 0, stream>>>(T, Mn);

        float* tmp = Mc; Mc = Mn; Mn = tmp;
    }
    final_site_kernel<<<1, 128, 0, stream>>>(Mc, out + koff[11]);
}